// Machine_79542794322000
// MI455X (gfx1250) — compile-verified
//
#include <hip/hip_runtime.h>
#include <hip/hip_bf16.h>

#define T_STEPS 256
#define BATCH 16
#define FDIM 256
#define NUM_BLOCKS 64
#define NPB 64
#define NTOT (NUM_BLOCKS * NPB)   // 4096
#define NNZB 512
#define SUBSTEPS 5
#define BN (BATCH * NTOT)         // 65536
#define TB (T_STEPS * BATCH)      // 4096

typedef __attribute__((ext_vector_type(2))) float v2f;
typedef __attribute__((ext_vector_type(8))) float v8f;

__device__ __forceinline__ v8f wmma_f32_16x16x4(v2f a, v2f b, v8f c) {
  // D = A(16x4 f32) * B(4x16 f32) + C(16x16 f32) on the CDNA5 matrix pipe.
  return __builtin_amdgcn_wmma_f32_16x16x4_f32(false, a, false, b, (short)0, c,
                                               false, false);
}

// ---------------------------------------------------------------------------
// Setup: zero the device-barrier counter (graph replays must start from 0).
// ---------------------------------------------------------------------------
__global__ void init_ctr(unsigned* ctr) {
  if (threadIdx.x == 0 && blockIdx.x == 0) *ctr = 0u;
}

// ---------------------------------------------------------------------------
// Setup: pre-swizzle BSR weight blocks into WMMA A-fragment lane order so the
// hot loop does one coalesced b64 load per lane per WMMA.
// Wp[(k*4 + wave)*16 + kk][lane] = float2{ W[k][m][kd], W[k][m][kd+1] }
//   m  = wave*16 + (lane&15)            (A-matrix row, lanes 0-15 / 16-31)
//   kd = kk*4 + (lane>>4)*2             (A-matrix K pair per 16x16x4 layout)
// ---------------------------------------------------------------------------
__global__ __launch_bounds__(128) void prep_w(const float* __restrict__ wv,
                                              float* __restrict__ wp) {
  const int blk = blockIdx.x;
  const int wave = threadIdx.x >> 5, lane = threadIdx.x & 31;
  const int m = wave * 16 + (lane & 15);
  const int hi = lane >> 4;
  const float2* src = (const float2*)(wv + ((size_t)blk * NPB + m) * NPB);
  float2* dst = (float2*)wp + (((size_t)blk * 4 + wave) * 16) * 32 + lane;
#pragma unroll
  for (int kk = 0; kk < 16; ++kk) {
    dst[kk * 32] = src[kk * 2 + hi];
  }
}

// ---------------------------------------------------------------------------
// Phase 1: drive[t,b,n] = gain[n] * sum_f ext[t,b,f]*w_in[n,f] + bias[n]
// GEMM (TB x F) x (F x N) with V_WMMA_F32_16X16X4_F32.
// Block: 128 thr (4 waves). M-tile 16 rows shared via LDS; wave w owns N-tile
// n0 + w*16.
// ---------------------------------------------------------------------------
__global__ __launch_bounds__(128) void proj_kernel(
    const float* __restrict__ ext, const float* __restrict__ w_in,
    const float* __restrict__ gain, const float* __restrict__ bias,
    float* __restrict__ drive) {
  __shared__ float aLds[16 * 260];  // +4 pad: conflict-free fragment reads
  const int m0 = blockIdx.x * 16;
  const int n0 = blockIdx.y * 64;
  const int tid = threadIdx.x;

  // Cooperative load of the 16x256 A tile (ext rows m0..m0+15).
#pragma unroll
  for (int e = 0; e < 32; ++e) {
    int i = tid + e * 128;
    int row = i >> 8, col = i & 255;
    aLds[row * 260 + col] = ext[(size_t)(m0 + row) * FDIM + col];
  }
  __syncthreads();

  const int wave = tid >> 5, lane = tid & 31;
  const int lo16 = lane & 15, hi = lane >> 4;
  const int n = n0 + wave * 16 + lo16;
  const float* wrow = w_in + (size_t)n * FDIM;  // B[k][n] = w_in[n][k]

  v8f acc = {};
#pragma unroll 4
  for (int kk = 0; kk < 64; ++kk) {
    const int kb = kk * 4 + hi * 2;
    v2f a = *(const v2f*)&aLds[lo16 * 260 + kb];
    v2f b = *(const v2f*)&wrow[kb];
    acc = wmma_f32_16x16x4(a, b, acc);
  }

  const float g = gain[n], bi = bias[n];
#pragma unroll
  for (int v = 0; v < 8; ++v) {
    int m = m0 + v + 8 * hi;  // C/D layout: VGPR v -> row v (+8 for hi lanes)
    drive[(size_t)m * NTOT + n] = g * acc[v] + bi;
  }
}

// ---------------------------------------------------------------------------
// Phase 2: persistent recurrent kernel. One workgroup per BSR row-block r.
// 1280 substeps; device-wide sync via atomic counter between substeps.
// Per substep: y(64x16) = sum_{k: rows[k]==r} W_k (64x64) * s[:,cols[k]] via
// f32 WMMA, then s_new = s*df + (1-df)*tanh(y + drive). e/a states are dead
// code w.r.t. the output trajectory and are skipped.
// ---------------------------------------------------------------------------
__global__ __launch_bounds__(128) void recur_persist(
    const float* __restrict__ s0, const float* __restrict__ drive_all,
    const float* __restrict__ wp, const int* __restrict__ rows,
    const int* __restrict__ cols, float* __restrict__ b0,
    float* __restrict__ b1, float* __restrict__ out,
    unsigned* __restrict__ ctr) {
  __shared__ float scol[2][16 * 68];  // double-buffered gathered col block

  const int r = blockIdx.x;
  const int tid = threadIdx.x;
  const int wave = tid >> 5, lane = tid & 31;
  const int lo16 = lane & 15, hi = lane >> 4;

  // [lo,hiK) = nnz range of sorted row index r (binary search, uniform).
  int lo, hiK;
  {
    int a = 0, b = NNZB;
    while (a < b) { int m = (a + b) >> 1; if (rows[m] < r) a = m + 1; else b = m; }
    lo = a;
    b = NNZB;
    while (a < b) { int m = (a + b) >> 1; if (rows[m] <= r) a = m + 1; else b = m; }
    hiK = a;
  }

  const float DF = 0.81873075307798182f;   // exp(-DT/TAU_FAST) = exp(-0.2)
  const float OMDF = 1.0f - DF;

  const float* sin = s0;
  unsigned step = 0;

  for (int t = 0; t < T_STEPS; ++t) {
    const float* drv = drive_all + (size_t)t * BN;
    for (int sub = 0; sub < SUBSTEPS; ++sub) {
      // Buffer chain: b0,b1,b0,b1, then the trajectory slot itself (next
      // step's sub0 reads from out + t*BN — no extra copy).
      float* sout = (sub == SUBSTEPS - 1) ? (out + (size_t)t * BN)
                                          : ((sub & 1) ? b1 : b0);

      v8f acc = {};
      if (lo < hiK) {  // preload first gathered column block into buffer 0
        const float* sp = sin + cols[lo] * NPB;
#pragma unroll
        for (int e = 0; e < 8; ++e) {
          int i = tid + e * 128;
          int bb = i >> 6, j = i & 63;
          scol[0][bb * 68 + j] = sp[(size_t)bb * NTOT + j];
        }
      }
      __syncthreads();

      for (int k = lo; k < hiK; ++k) {
        const int cur = (k - lo) & 1;
        if (k + 1 < hiK) {  // overlap next gather with this block's WMMAs
          const float* sp = sin + cols[k + 1] * NPB;
#pragma unroll
          for (int e = 0; e < 8; ++e) {
            int i = tid + e * 128;
            int bb = i >> 6, j = i & 63;
            scol[cur ^ 1][bb * 68 + j] = sp[(size_t)bb * NTOT + j];
          }
        }
        const v2f* wpk =
            (const v2f*)wp + (((size_t)k * 4 + wave) * 16) * 32 + lane;
#pragma unroll
        for (int kk = 0; kk < 16; ++kk) {
          v2f a = wpk[kk * 32];  // pre-swizzled A fragment, coalesced b64
          const int kb = kk * 4 + hi * 2;
          v2f b = *(const v2f*)&scol[cur][lo16 * 68 + kb];
          acc = wmma_f32_16x16x4(a, b, acc);
        }
        __syncthreads();
      }

      // Epilogue: this lane owns 8 consecutive neurons for batch lo16.
      {
        const int nb = r * NPB + wave * 16 + 8 * hi;
        const size_t base = (size_t)lo16 * NTOT + nb;
        const float4* so = (const float4*)(sin + base);
        const float4* dv = (const float4*)(drv + base);
        float4 sA = so[0], sB = so[1];
        float4 dA = dv[0], dB = dv[1];
        float sold[8] = {sA.x, sA.y, sA.z, sA.w, sB.x, sB.y, sB.z, sB.w};
        float dvv[8] = {dA.x, dA.y, dA.z, dA.w, dB.x, dB.y, dB.z, dB.w};
        float res[8];
#pragma unroll
        for (int v = 0; v < 8; ++v) {
          res[v] = sold[v] * DF + OMDF * tanhf(acc[v] + dvv[v]);
        }
        float4* op = (float4*)(sout + base);
        op[0] = make_float4(res[0], res[1], res[2], res[3]);
        op[1] = make_float4(res[4], res[5], res[6], res[7]);
      }

      // Device-wide barrier: all 64 workgroups (co-resident) arrive.
      ++step;
      __threadfence();
      __syncthreads();
      if (tid == 0) {
        __hip_atomic_fetch_add(ctr, 1u, __ATOMIC_ACQ_REL,
                               __HIP_MEMORY_SCOPE_AGENT);
        const unsigned target = (unsigned)NUM_BLOCKS * step;
        while (__hip_atomic_load(ctr, __ATOMIC_ACQUIRE,
                                 __HIP_MEMORY_SCOPE_AGENT) < target) {
          __builtin_amdgcn_s_sleep(8);
        }
      }
      __syncthreads();
      sin = sout;
    }
  }
}

// ---------------------------------------------------------------------------
extern "C" void kernel_launch(void* const* d_in, const int* in_sizes, int n_in,
                              void* d_out, int out_size, void* d_ws,
                              size_t ws_size, hipStream_t stream) {
  const float* ext  = (const float*)d_in[0];  // (T,B,F)
  const float* w_in = (const float*)d_in[1];  // (N,F)
  const float* gain = (const float*)d_in[2];  // (N,)
  const float* bias = (const float*)d_in[3];  // (N,)
  const float* wv   = (const float*)d_in[4];  // (NNZ,64,64)
  const float* s0   = (const float*)d_in[5];  // (B,N)
  const int* rows   = (const int*)d_in[8];    // (NNZ,) sorted
  const int* cols   = (const int*)d_in[9];    // (NNZ,)
  float* out = (float*)d_out;                 // (T,B,N)

  float* ws = (float*)d_ws;
  float* drive = ws;                                   // T*B*N floats (67MB)
  float* wp = drive + (size_t)T_STEPS * BN;            // NNZ*64*64 (8.4MB)
  float* b0 = wp + (size_t)NNZB * NPB * NPB;           // B*N
  float* b1 = b0 + BN;                                 // B*N
  unsigned* ctr = (unsigned*)(b1 + BN);                // barrier counter

  init_ctr<<<1, 32, 0, stream>>>(ctr);
  prep_w<<<NNZB, 128, 0, stream>>>(wv, wp);
  dim3 pg(TB / 16, NTOT / 64);
  proj_kernel<<<pg, 128, 0, stream>>>(ext, w_in, gain, bias, drive);
  recur_persist<<<NUM_BLOCKS, 128, 0, stream>>>(s0, drive, wp, rows, cols, b0,
                                                b1, out, ctr);
}